// craft_mae_loss_22436909154406
// MI455X (gfx1250) — compile-verified
//
#include <hip/hip_runtime.h>

// CRAFT MAE loss for MI455X (gfx1250, wave32).
// B=32, H=W=512. Memory-bound two-pass OHEM reduction.
// bg_mask == 1 - fg_mask exactly, so bg_mask is never loaded (6 tensors, not 7).

typedef float v2f __attribute__((ext_vector_type(2)));
typedef float v8f __attribute__((ext_vector_type(8)));

#define HW4        65536    // (512*512)/4 float4 per batch
#define THREADS    256      // 8 wave32 per block
#define ITER       4
#define BLK_PER_B  64       // 64 blocks * 256 thr * 4 f4 = 65536 f4 per batch
#define NBATCH     32
#define NBLK       (NBATCH * BLK_PER_B)   // 2048 blocks per pass
#define CONF_TH    0.5f

// ---------------------------------------------------------------- init ------
__global__ void init_ws(unsigned int* __restrict__ thresh) {
  if (threadIdx.x < NBATCH) thresh[threadIdx.x] = 0u;  // 0u == bits of +0.0f
}

// ------------------------------------------------------------ pass A --------
// Per-batch max of neg_loss (u32 atomic max on non-negative float bits) and
// fg-masked partial sums of l_total*conf and conf.
__device__ __forceinline__ void comp_a(float rtv, float atv, float rpv, float apv,
                                       float cv, float fgv,
                                       float& s_l, float& s_c, float& mx) {
  float conf = (cv >= CONF_TH) ? cv : 0.0f;
  float l    = fabsf(rtv - rpv) + fabsf(atv - apv);
  float lc   = l * conf;
  s_l += lc * fgv;            // fg in {0,1}
  s_c += conf * fgv;
  mx = fmaxf(mx, lc * (1.0f - fgv));   // neg_loss = l_total * bg
}

__global__ __launch_bounds__(THREADS)
void pass_a(const float4* __restrict__ rt, const float4* __restrict__ at,
            const float4* __restrict__ rp, const float4* __restrict__ ap,
            const float4* __restrict__ cf, const float4* __restrict__ fg,
            unsigned int* __restrict__ thresh,
            float* __restrict__ pnum, float* __restrict__ pden) {
  const int b    = blockIdx.y;
  const int tid  = threadIdx.x;
  const int base = b * HW4 + blockIdx.x * (THREADS * ITER) + tid;
  float s_l = 0.f, s_c = 0.f, mx = 0.f;
#pragma unroll
  for (int it = 0; it < ITER; ++it) {
    const int i = base + it * THREADS;
    if (it + 1 < ITER) {                 // gfx1250 global_prefetch_b8
      __builtin_prefetch(rt + i + THREADS, 0, 3);
      __builtin_prefetch(at + i + THREADS, 0, 3);
      __builtin_prefetch(rp + i + THREADS, 0, 3);
      __builtin_prefetch(ap + i + THREADS, 0, 3);
      __builtin_prefetch(cf + i + THREADS, 0, 3);
      __builtin_prefetch(fg + i + THREADS, 0, 3);
    }
    float4 a = rt[i], t = at[i], e = rp[i], g = ap[i], h = cf[i], f = fg[i];
    comp_a(a.x, t.x, e.x, g.x, h.x, f.x, s_l, s_c, mx);
    comp_a(a.y, t.y, e.y, g.y, h.y, f.y, s_l, s_c, mx);
    comp_a(a.z, t.z, e.z, g.z, h.z, f.z, s_l, s_c, mx);
    comp_a(a.w, t.w, e.w, g.w, h.w, f.w, s_l, s_c, mx);
  }
  // wave32 tree reduction
#pragma unroll
  for (int o = 16; o > 0; o >>= 1) {
    s_l += __shfl_xor(s_l, o, 32);
    s_c += __shfl_xor(s_c, o, 32);
    mx   = fmaxf(mx, __shfl_xor(mx, o, 32));
  }
  __shared__ float sm_l[8], sm_c[8], sm_m[8];
  const int wid = tid >> 5, lane = tid & 31;
  if (lane == 0) { sm_l[wid] = s_l; sm_c[wid] = s_c; sm_m[wid] = mx; }
  __syncthreads();
  if (tid == 0) {
    float tl = 0.f, tc = 0.f, tm = 0.f;
#pragma unroll
    for (int w = 0; w < 8; ++w) {
      tl += sm_l[w]; tc += sm_c[w]; tm = fmaxf(tm, sm_m[w]);
    }
    const int blk = b * gridDim.x + blockIdx.x;
    pnum[blk] = tl;
    pden[blk] = tc;
    atomicMax(&thresh[b], __float_as_uint(tm));  // exact: values >= 0
  }
}

// ------------------------------------------------------------ pass B --------
// Reverse traversal so the head of this pass hits the tail of pass A's stream
// still resident in the 192MB L2 (working set = 201MB).
__global__ __launch_bounds__(THREADS)
void pass_b(const float4* __restrict__ rt, const float4* __restrict__ at,
            const float4* __restrict__ rp, const float4* __restrict__ ap,
            const float4* __restrict__ cf, const float4* __restrict__ fg,
            const unsigned int* __restrict__ thresh,
            float* __restrict__ pnum, float* __restrict__ pden) {
  const int b    = (NBATCH - 1) - blockIdx.y;            // reversed
  const int bx   = (BLK_PER_B - 1) - blockIdx.x;         // reversed
  const int tid  = threadIdx.x;
  const int base = b * HW4 + bx * (THREADS * ITER) + tid;
  const float th = __uint_as_float(thresh[b]);           // uniform scalar load
  float s_l = 0.f, s_c = 0.f;
#pragma unroll
  for (int it = 0; it < ITER; ++it) {
    const int i = base + it * THREADS;
    float4 a = rt[i], t = at[i], e = rp[i], g = ap[i], h = cf[i], f = fg[i];
    const float rv[4] = {a.x, a.y, a.z, a.w};
    const float tv[4] = {t.x, t.y, t.z, t.w};
    const float ev[4] = {e.x, e.y, e.z, e.w};
    const float gv[4] = {g.x, g.y, g.z, g.w};
    const float hv[4] = {h.x, h.y, h.z, h.w};
    const float fv[4] = {f.x, f.y, f.z, f.w};
#pragma unroll
    for (int k = 0; k < 4; ++k) {
      float conf = (hv[k] >= CONF_TH) ? hv[k] : 0.0f;
      float lc   = (fabsf(rv[k] - ev[k]) + fabsf(tv[k] - gv[k])) * conf;
      // bg>0 <=> fg==0; recomputation is bit-identical to pass A, so the
      // argmax element satisfies lc >= th exactly.
      float sel = ((fv[k] == 0.0f) && (lc >= th)) ? 1.0f : 0.0f;
      s_l += lc * sel;
      s_c += conf * sel;
    }
  }
#pragma unroll
  for (int o = 16; o > 0; o >>= 1) {
    s_l += __shfl_xor(s_l, o, 32);
    s_c += __shfl_xor(s_c, o, 32);
  }
  __shared__ float sm_l[8], sm_c[8];
  const int wid = tid >> 5, lane = tid & 31;
  if (lane == 0) { sm_l[wid] = s_l; sm_c[wid] = s_c; }
  __syncthreads();
  if (tid == 0) {
    float tl = 0.f, tc = 0.f;
#pragma unroll
    for (int w = 0; w < 8; ++w) { tl += sm_l[w]; tc += sm_c[w]; }
    const int blk = b * gridDim.x + bx;
    pnum[blk] = tl;
    pden[blk] = tc;
  }
}

// ----------------------------------------------------------- finalize -------
// One wave32 (EXEC all-1s, as WMMA requires). Each lane accumulates a strided
// slice of the 2*NBLK partials; the 32 lane accumulators are then reduced
// through the matrix unit: D = A x ones, V_WMMA_F32_16X16X4_F32 rowsums.
__global__ void finalize(const float* __restrict__ pnum,
                         const float* __restrict__ pden,
                         float* __restrict__ out) {
  const int lane = threadIdx.x;  // launched with exactly 32 threads
  float n = 0.f, d = 0.f;
  for (int i = lane; i < 2 * NBLK; i += 32) { n += pnum[i]; d += pden[i]; }

  // A layout (16x4 f32, 2 VGPRs): lanes 0-15 hold K=0/1, lanes 16-31 K=2/3.
  // Put each lane's accumulator in slot .x, zero in .y: row m sums
  // lane_m + lane_{16+m}. B = all-ones => D[m][n] = rowsum_m.
  v2f an; an.x = n;   an.y = 0.f;
  v2f ad; ad.x = d;   ad.y = 0.f;
  v2f ob; ob.x = 1.f; ob.y = 1.f;
  v8f c0 = {};
  v8f dn = __builtin_amdgcn_wmma_f32_16x16x4_f32(false, an, false, ob,
                                                 (short)0, c0, false, false);
  v8f dd = __builtin_amdgcn_wmma_f32_16x16x4_f32(false, ad, false, ob,
                                                 (short)0, c0, false, false);
  // Lane L<16 holds D[0..7][L] in its 8 VGPRs; lane L>=16 holds D[8..15][L-16].
  float pn = dn[0]+dn[1]+dn[2]+dn[3]+dn[4]+dn[5]+dn[6]+dn[7];
  float pd = dd[0]+dd[1]+dd[2]+dd[3]+dd[4]+dd[5]+dd[6]+dd[7];
  pn += __shfl_xor(pn, 16, 32);   // rows 0-7 + rows 8-15 => full sum
  pd += __shfl_xor(pd, 16, 32);
  if (lane == 0) out[0] = pn / (pd + 1e-7f);
}

// ------------------------------------------------------------- launch -------
extern "C" void kernel_launch(void* const* d_in, const int* in_sizes, int n_in,
                              void* d_out, int out_size, void* d_ws, size_t ws_size,
                              hipStream_t stream) {
  (void)in_sizes; (void)n_in; (void)out_size; (void)ws_size;
  const float4* rt = (const float4*)d_in[0];  // region_true
  const float4* at = (const float4*)d_in[1];  // affinity_true
  const float4* rp = (const float4*)d_in[2];  // region_pred
  const float4* ap = (const float4*)d_in[3];  // affinity_pred
  const float4* cf = (const float4*)d_in[4];  // confidence
  const float4* fg = (const float4*)d_in[5];  // fg_mask  (bg = 1 - fg, unused)

  // ws layout: [0,128) u32 thresh[32]; then pnum[2*NBLK]; then pden[2*NBLK].
  unsigned int* thresh = (unsigned int*)d_ws;
  float* pnum = (float*)((char*)d_ws + 128);
  float* pden = pnum + 2 * NBLK;

  init_ws<<<1, 32, 0, stream>>>(thresh);
  dim3 grid(BLK_PER_B, NBATCH);
  pass_a<<<grid, THREADS, 0, stream>>>(rt, at, rp, ap, cf, fg,
                                       thresh, pnum, pden);
  pass_b<<<grid, THREADS, 0, stream>>>(rt, at, rp, ap, cf, fg,
                                       thresh, pnum + NBLK, pden + NBLK);
  finalize<<<1, 32, 0, stream>>>(pnum, pden, (float*)d_out);
}